// Nalui2Layer_63367947485910
// MI455X (gfx1250) — compile-verified
//
#include <hip/hip_runtime.h>
#include <math.h>

#define BATCH 4096
#define NX 256
#define NY 128
#define MROWS 32              // batch rows per block (2 WMMA M-tiles per wave)

typedef __attribute__((ext_vector_type(2))) float v2f;
typedef __attribute__((ext_vector_type(8))) float v8f;

__device__ __forceinline__ float sigmoid_f(float v) {
    return 1.0f / (1.0f + expf(-v));
}

// Swizzle a K index (0..255) into the packed per-row stream order:
// hi-half h = (k>>1)&1 gets its own contiguous 128-float stream, within which
// k-step t = k>>2 contributes 2 consecutive floats. A lane (fixed h) then
// reads a contiguous stream: one float4 = two k-steps.
__device__ __host__ __forceinline__ int pack_k(int k) {
    return (((k >> 1) & 1) << 7) + ((k >> 2) << 1) + (k & 1);
}

// ---------------------------------------------------------------------------
// Prep kernel: gate the weights, store the four B matrices transposed
// (ny-major) AND k-packed so WMMA B-fragment loads are contiguous float4.
// ---------------------------------------------------------------------------
__global__ void nalu_prep(const float* __restrict__ Wa, const float* __restrict__ Ma,
                          const float* __restrict__ Wm, const float* __restrict__ Mm,
                          float* __restrict__ Badd, float* __restrict__ Bmul,
                          float* __restrict__ BL,   float* __restrict__ BP) {
    int idx = blockIdx.x * blockDim.x + threadIdx.x;
    if (idx >= NX * NY) return;
    int j = idx >> 8;          // ny index
    int i = idx & (NX - 1);    // nx (K) index
    int src = i * NY + j;      // (nx,ny) row-major source
    int dst = j * NX + pack_k(i);

    Badd[dst] = tanhf(Wa[src]) * sigmoid_f(Ma[src]);
    Bmul[dst] = tanhf(Wm[src]) * sigmoid_f(Mm[src]);

    // Wf[j,i] is the flat (nx,ny) row-major buffer read at flat index j*NX+i == idx.
    float v = tanhf(Wm[idx]) * sigmoid_f(Mm[idx]);
    float t = 1.0f - 2.0f * fabsf(v);
    BL[dst] = logf(fabsf(t));            // -inf when t==0 -> exp -> 0, correct
    BP[dst] = (t < 0.0f) ? 1.0f : 0.0f;
}

// ---------------------------------------------------------------------------
// Main kernel: 4 fused fp32 WMMA GEMMs (a1, log-m1, log|ms1|, parity(ms1)).
// Block = 256 threads = 8 waves; block owns batch rows [32b,32b+32) (two
// 16-row M-tiles), wave w owns ny columns [16w,16w+16). Each B fragment is
// loaded once per K-step and reused across both M-tiles (halves L2 traffic).
// A-tiles staged once per block into LDS in WMMA lane order.
// ---------------------------------------------------------------------------
__global__ __launch_bounds__(256) void nalu_main(
    const float* __restrict__ x, const float* __restrict__ g,
    const float* __restrict__ Badd, const float* __restrict__ Bmul,
    const float* __restrict__ BL,   const float* __restrict__ BP,
    float* __restrict__ out) {

    __shared__ float sX[MROWS * NX];   // raw x tile          (packed)
    __shared__ float sL[MROWS * NX];   // log(max(|x|,1e-7))  (packed)
    __shared__ float sN[MROWS * NX];   // x<0 ? 1 : 0         (packed)

    const int btile = blockIdx.x;      // 0..127

    // Cooperative staging: coalesced global reads, transforms computed ONCE
    // per block, stored pre-swizzled for fragment loads.
    for (int e = threadIdx.x; e < MROWS * NX; e += 256) {
        int r = e >> 8;                // tile row 0..31
        int k = e & (NX - 1);          // K index
        float xv = x[(btile * MROWS + r) * NX + k];
        int p = r * NX + pack_k(k);
        sX[p] = xv;
        sL[p] = __logf(fmaxf(fabsf(xv), 1e-7f));
        sN[p] = (xv < 0.0f) ? 1.0f : 0.0f;
    }
    __syncthreads();

    const int lane = threadIdx.x & 31;
    const int wave = threadIdx.x >> 5;
    const int row  = lane & 15;        // M (A side) / N (B,D side)
    const int hi   = lane >> 4;        // K pair select

    const int aoff0 = row * NX + (hi << 7);          // M-tile 0 packed stream
    const int aoff1 = (16 + row) * NX + (hi << 7);   // M-tile 1 packed stream
    const int jcol = wave * 16 + row;
    const int boff = jcol * NX + (hi << 7);          // lane's packed B stream
    const float* pBa = Badd + boff;
    const float* pBm = Bmul + boff;
    const float* pBl = BL   + boff;
    const float* pBp = BP   + boff;

    v8f accA0 = {}, accA1 = {};   // a1  = x @ WM_add
    v8f accM0 = {}, accM1 = {};   // log-domain m1 = log|x| @ WM_mul
    v8f accL0 = {}, accL1 = {};   // log|ms1| = negmask @ log|1-2Wf|
    v8f accP0 = {}, accP1 = {};   // parity count = negmask @ [1-2Wf<0]

    #pragma unroll 2
    for (int s = 0; s < 64; s += 2) {            // two 16x16x4 k-steps per iter
        const int o = s << 1;                    // float offset in packed stream
        float4 ba = *(const float4*)(pBa + o);
        float4 bm = *(const float4*)(pBm + o);
        float4 bl = *(const float4*)(pBl + o);
        float4 bp = *(const float4*)(pBp + o);
        float4 ax0 = *(const float4*)(sX + aoff0 + o);
        float4 al0 = *(const float4*)(sL + aoff0 + o);
        float4 an0 = *(const float4*)(sN + aoff0 + o);
        float4 ax1 = *(const float4*)(sX + aoff1 + o);
        float4 al1 = *(const float4*)(sL + aoff1 + o);
        float4 an1 = *(const float4*)(sN + aoff1 + o);

        v2f bA0 = { ba.x, ba.y }, bA1 = { ba.z, ba.w };
        v2f bM0 = { bm.x, bm.y }, bM1 = { bm.z, bm.w };
        v2f bL0 = { bl.x, bl.y }, bL1 = { bl.z, bl.w };
        v2f bP0 = { bp.x, bp.y }, bP1 = { bp.z, bp.w };

        v2f a;
        a = (v2f){ ax0.x, ax0.y };
        accA0 = __builtin_amdgcn_wmma_f32_16x16x4_f32(false, a, false, bA0, (short)0, accA0, false, false);
        a = (v2f){ ax0.z, ax0.w };
        accA0 = __builtin_amdgcn_wmma_f32_16x16x4_f32(false, a, false, bA1, (short)0, accA0, false, false);
        a = (v2f){ ax1.x, ax1.y };
        accA1 = __builtin_amdgcn_wmma_f32_16x16x4_f32(false, a, false, bA0, (short)0, accA1, false, false);
        a = (v2f){ ax1.z, ax1.w };
        accA1 = __builtin_amdgcn_wmma_f32_16x16x4_f32(false, a, false, bA1, (short)0, accA1, false, false);

        a = (v2f){ al0.x, al0.y };
        accM0 = __builtin_amdgcn_wmma_f32_16x16x4_f32(false, a, false, bM0, (short)0, accM0, false, false);
        a = (v2f){ al0.z, al0.w };
        accM0 = __builtin_amdgcn_wmma_f32_16x16x4_f32(false, a, false, bM1, (short)0, accM0, false, false);
        a = (v2f){ al1.x, al1.y };
        accM1 = __builtin_amdgcn_wmma_f32_16x16x4_f32(false, a, false, bM0, (short)0, accM1, false, false);
        a = (v2f){ al1.z, al1.w };
        accM1 = __builtin_amdgcn_wmma_f32_16x16x4_f32(false, a, false, bM1, (short)0, accM1, false, false);

        a = (v2f){ an0.x, an0.y };
        accL0 = __builtin_amdgcn_wmma_f32_16x16x4_f32(false, a, false, bL0, (short)0, accL0, false, false);
        accP0 = __builtin_amdgcn_wmma_f32_16x16x4_f32(false, a, false, bP0, (short)0, accP0, false, false);
        a = (v2f){ an0.z, an0.w };
        accL0 = __builtin_amdgcn_wmma_f32_16x16x4_f32(false, a, false, bL1, (short)0, accL0, false, false);
        accP0 = __builtin_amdgcn_wmma_f32_16x16x4_f32(false, a, false, bP1, (short)0, accP0, false, false);
        a = (v2f){ an1.x, an1.y };
        accL1 = __builtin_amdgcn_wmma_f32_16x16x4_f32(false, a, false, bL0, (short)0, accL1, false, false);
        accP1 = __builtin_amdgcn_wmma_f32_16x16x4_f32(false, a, false, bP0, (short)0, accP1, false, false);
        a = (v2f){ an1.z, an1.w };
        accL1 = __builtin_amdgcn_wmma_f32_16x16x4_f32(false, a, false, bL1, (short)0, accL1, false, false);
        accP1 = __builtin_amdgcn_wmma_f32_16x16x4_f32(false, a, false, bP1, (short)0, accP1, false, false);
    }

    // Epilogue. D layout: lanes 0-15: VGPR r = D[M=r][N=lane]; lanes 16-31: M=r+8.
    const float g1 = sigmoid_f(g[jcol]);
    const int bbase = btile * MROWS + hi * 8;
    #pragma unroll
    for (int r = 0; r < 8; ++r) {
        // M-tile 0
        {
            float m1  = __expf(fminf(accM0[r], 20.0f));
            float mag = __expf(accL0[r]);                     // |ms1|
            int   c   = (int)accP0[r];                        // exact integer
            float par = (c & 1) ? -1.0f : 1.0f;
            float ms1 = fminf(fmaxf(par * mag, -1.0f), 1.0f);
            float o   = g1 * accA0[r] + (1.0f - g1) * m1 * ms1;
            __builtin_nontemporal_store(o, &out[(bbase + r) * NY + jcol]);
        }
        // M-tile 1
        {
            float m1  = __expf(fminf(accM1[r], 20.0f));
            float mag = __expf(accL1[r]);
            int   c   = (int)accP1[r];
            float par = (c & 1) ? -1.0f : 1.0f;
            float ms1 = fminf(fmaxf(par * mag, -1.0f), 1.0f);
            float o   = g1 * accA1[r] + (1.0f - g1) * m1 * ms1;
            __builtin_nontemporal_store(o, &out[(bbase + 16 + r) * NY + jcol]);
        }
    }
}

extern "C" void kernel_launch(void* const* d_in, const int* in_sizes, int n_in,
                              void* d_out, int out_size, void* d_ws, size_t ws_size,
                              hipStream_t stream) {
    const float* x  = (const float*)d_in[0];
    const float* Wa = (const float*)d_in[1];
    const float* Ma = (const float*)d_in[2];
    const float* Wm = (const float*)d_in[3];
    const float* Mm = (const float*)d_in[4];
    const float* g  = (const float*)d_in[5];
    float* out = (float*)d_out;

    float* ws   = (float*)d_ws;          // 4 * 32768 floats = 512 KB
    float* Badd = ws;
    float* Bmul = ws + 1 * NX * NY;
    float* BL   = ws + 2 * NX * NY;
    float* BP   = ws + 3 * NX * NY;

    nalu_prep<<<(NX * NY + 255) / 256, 256, 0, stream>>>(Wa, Ma, Wm, Mm, Badd, Bmul, BL, BP);
    nalu_main<<<BATCH / MROWS, 256, 0, stream>>>(x, g, Badd, Bmul, BL, BP, out);
}